// Lattice_71287867179278
// MI455X (gfx1250) — compile-verified
//
#include <hip/hip_runtime.h>

typedef __attribute__((ext_vector_type(2))) float v2f;
typedef __attribute__((ext_vector_type(8))) float v8f;

#define SOM_M 65536
#define SOM_B 2048
#define SOM_D 32
#define M_SLICES 32                              // waves slicing the M dim
#define TILES_PER_WAVE (SOM_M / M_SLICES / 16)   // 128
// Per-row-constant bias keeps argmin identical; one global constant >
// max ||x_b||^2 (chi^2_32, max over 2048 draws ~= 63) makes dist' > 0 so the
// raw f32 bit pattern is already a monotonic unsigned sort key.
#define DIST_BIAS 256.0f

// ---------------------------------------------------------------------------
// Kernel 1: wnorm[m] = ||w_m||^2, and reset per-row best keys to ~0ull.
// ---------------------------------------------------------------------------
__global__ __launch_bounds__(256) void som_prep_kernel(
    const float* __restrict__ w, float* __restrict__ wnorm,
    unsigned long long* __restrict__ best) {
  const int tid = blockIdx.x * blockDim.x + threadIdx.x;  // 0..65535
  if (tid < SOM_M) {
    const float* wr = w + tid * SOM_D;
    float s = 0.0f;
#pragma unroll
    for (int k = 0; k < SOM_D; k += 4) {
      float4 v = *(const float4*)(wr + k);
      s += v.x * v.x + v.y * v.y + v.z * v.z + v.w * v.w;
    }
    wnorm[tid] = s;
  }
  if (tid < SOM_B) best[tid] = ~0ull;
}

// ---------------------------------------------------------------------------
// Kernel 2: BMU search. One wave = one 16-row B tile x one M slice, using
// v_wmma_f32_16x16x4_f32 (K stepped 8x over D=32, exact f32 accumulate).
//
// f32 16x16 C/D layout (wave32): lane L<16 holds column N=L, rows 0..7 in
// c[0..7]; lane L>=16 holds column N=L-16, rows 8..15 in c[0..7].
// A (16x4): lane L: row M=L%16, K = {0,1} (lanes 0-15) / {2,3} (lanes 16-31).
//
// Per element the argmin update is v_fma + v_max + v_min_u64; all 8 B
// fragments are preloaded per tile so the loads form one clause and overlap
// the WMMA chain instead of serializing load->wait->wmma.
// ---------------------------------------------------------------------------
__global__ __launch_bounds__(256) void som_bmu_kernel(
    const float* __restrict__ x,      // [B, D]
    const float* __restrict__ w,      // [M, D]
    const float* __restrict__ wnorm,  // [M]
    unsigned long long* __restrict__ best) {  // [B]
  const int lane   = threadIdx.x & 31;
  const int waveId = blockIdx.x * 8 + (threadIdx.x >> 5);  // 0..4095
  const int bTile  = waveId >> 5;         // 0..127 (16 rows of x each)
  const int mSlice = waveId & (M_SLICES - 1);
  const int hi  = lane >> 4;              // 0 or 1 (lane half)
  const int l16 = lane & 15;

  // Preload A fragments; reused across all tiles of this wave.
  const float* xrow = x + (bTile * 16 + l16) * SOM_D;
  v2f A[8];
#pragma unroll
  for (int s = 0; s < 8; ++s)
    A[s] = *(const v2f*)(xrow + 4 * s + 2 * hi);

  unsigned long long bestk[8];
#pragma unroll
  for (int r = 0; r < 8; ++r) bestk[r] = ~0ull;

  const int mBase = mSlice * (SOM_M / M_SLICES);
  for (int t = 0; t < TILES_PER_WAVE; ++t) {
    const int mcol = mBase + t * 16 + l16;   // this lane's codebook column
    const float* wcol = w + mcol * SOM_D;
    if (t < TILES_PER_WAVE - 1)
      __builtin_prefetch(wcol + SOM_D * 16, 0, 3);  // next tile of w

    // One clause: wnorm + all 8 B fragments in flight together.
    const float wn = wnorm[mcol];
    v2f Bf[8];
#pragma unroll
    for (int s = 0; s < 8; ++s)
      Bf[s] = *(const v2f*)(wcol + 4 * s + 2 * hi);

    v8f c = {};
#pragma unroll
    for (int s = 0; s < 8; ++s)
      c = __builtin_amdgcn_wmma_f32_16x16x4_f32(
          /*neg_a=*/false, A[s], /*neg_b=*/false, Bf[s],
          /*c_mod=*/(short)0, c, /*reuse_a=*/false, /*reuse_b=*/false);

    const float wnb = wn + DIST_BIAS;  // single add serves all 8 rows
#pragma unroll
    for (int r = 0; r < 8; ++r) {
      float dist = fmaf(-2.0f, c[r], wnb);  // > 0 by construction
      dist = fmaxf(dist, 0.0f);             // cheap safety clamp
      const unsigned long long k64 =
          ((unsigned long long)__float_as_uint(dist) << 32) |
          (unsigned int)mcol;
      bestk[r] = (k64 < bestk[r]) ? k64 : bestk[r];
    }
  }

  // Min-reduce across the 16 lanes that share the same 8 rows.
#pragma unroll
  for (int r = 0; r < 8; ++r) {
    unsigned long long v = bestk[r];
#pragma unroll
    for (int mask = 1; mask <= 8; mask <<= 1) {
      const unsigned long long o = __shfl_xor(v, mask, 32);
      v = (o < v) ? o : v;
    }
    bestk[r] = v;
  }

  if (l16 == 0) {
    const int rowBase = bTile * 16 + hi * 8;
#pragma unroll
    for (int r = 0; r < 8; ++r)
      atomicMin(&best[rowBase + r], bestk[r]);   // unique keys -> deterministic
  }
}

// ---------------------------------------------------------------------------
// Kernel 3: decode winning index, gather grid coordinates.
// ---------------------------------------------------------------------------
__global__ __launch_bounds__(256) void som_gather_kernel(
    const unsigned long long* __restrict__ best,
    const float* __restrict__ grid,  // [M, 2]
    float* __restrict__ out) {       // [B, 2]
  const int b = blockIdx.x * blockDim.x + threadIdx.x;
  if (b < SOM_B) {
    const unsigned int idx = (unsigned int)(best[b] & 0xFFFFFFFFull);
    out[2 * b + 0] = grid[2 * idx + 0];
    out[2 * b + 1] = grid[2 * idx + 1];
  }
}

extern "C" void kernel_launch(void* const* d_in, const int* in_sizes, int n_in,
                              void* d_out, int out_size, void* d_ws,
                              size_t ws_size, hipStream_t stream) {
  (void)in_sizes; (void)n_in; (void)out_size; (void)ws_size;
  const float* x    = (const float*)d_in[0];  // [2048, 32]
  const float* grid = (const float*)d_in[1];  // [65536, 2] (leading 1 dropped)
  const float* w    = (const float*)d_in[2];  // [65536, 32]
  float* out = (float*)d_out;                 // [2048, 2]

  unsigned long long* best = (unsigned long long*)d_ws;            // 16 KB
  float* wnorm = (float*)((char*)d_ws + SOM_B * sizeof(unsigned long long));

  som_prep_kernel<<<SOM_M / 256, 256, 0, stream>>>(w, wnorm, best);
  // 4096 waves = 128 b-tiles * 32 m-slices; 8 waves (one b-tile) per block.
  som_bmu_kernel<<<128 * M_SLICES / 8, 256, 0, stream>>>(x, w, wnorm, best);
  som_gather_kernel<<<SOM_B / 256, 256, 0, stream>>>(best, grid, out);
}